// SetAbstraction_83468394430484
// MI455X (gfx1250) — compile-verified
//
#include <hip/hip_runtime.h>
#include <hip/hip_bf16.h>

#define BQ 4
#define NQ 8192
#define NPOINT 2048
#define NSAMPLE 32
#define RADIUS 0.5f
#define IN_DIM 64
#define OUT_DIM 128
#define ED 512          // OUT_DIM * EXP
#define GROUPS 8
#define EPS 1e-5f
#define KPAD 68         // 67 padded to multiple of 4

typedef __attribute__((ext_vector_type(2))) float v2f;
typedef __attribute__((ext_vector_type(8))) float v8f;

union AccU { v8f v; float f[8]; };

__device__ __forceinline__ v8f wmma_step(const float* Arow, const float* Brow,
                                         int kk, int kh, v8f acc) {
    v2f a, b;
    a.x = Arow[kk + 2 * kh];
    a.y = Arow[kk + 2 * kh + 1];
    b.x = Brow[kk + 2 * kh];
    b.y = Brow[kk + 2 * kh + 1];
    return __builtin_amdgcn_wmma_f32_16x16x4_f32(false, a, false, b,
                                                 (short)0, acc, false, false);
}

__device__ __forceinline__ float silu(float x) {
    return x / (1.0f + __expf(-x));
}

// ---------------------------------------------------------------- init stats
__global__ void init_stats_kernel(float* statsAcc) {
    int i = blockIdx.x * blockDim.x + threadIdx.x;
    if (i < 3 * BQ * GROUPS * 2) statsAcc[i] = 0.0f;
}

// ---------------------------------------------------------------- FPS
// One block (1024 threads) per batch. Each thread owns 8 consecutive points
// in registers. Per step: broadcast centroid, update dmin, 2-level argmax.
__global__ __launch_bounds__(1024) void fps_kernel(const float* __restrict__ xyz,
                                                   float* __restrict__ newxyz_out) {
    const int b = blockIdx.x;
    const int tid = threadIdx.x;
    const int lane = tid & 31, wid = tid >> 5;
    const int base = tid * 8;

    float px[8], py[8], pz[8], dmin[8];
#pragma unroll
    for (int j = 0; j < 8; ++j) {
        px[j] = xyz[(b * 3 + 0) * NQ + base + j];
        py[j] = xyz[(b * 3 + 1) * NQ + base + j];
        pz[j] = xyz[(b * 3 + 2) * NQ + base + j];
        dmin[j] = 1e10f;
    }

    __shared__ float cb[3];
    __shared__ float rv[32];
    __shared__ int   ri[32];
    __shared__ int   far_sh;

    int far = 0;
    for (int it = 0; it < NPOINT; ++it) {
        if (tid == (far >> 3)) {
            int sl = far & 7;
            cb[0] = px[sl]; cb[1] = py[sl]; cb[2] = pz[sl];
        }
        __syncthreads();
        float cx = cb[0], cy = cb[1], cz = cb[2];
        if (tid == 0) {
            newxyz_out[(b * 3 + 0) * NPOINT + it] = cx;
            newxyz_out[(b * 3 + 1) * NPOINT + it] = cy;
            newxyz_out[(b * 3 + 2) * NPOINT + it] = cz;
        }
        float bv = -1.0f; int bi = 0;
#pragma unroll
        for (int j = 0; j < 8; ++j) {
            float dx = px[j] - cx, dy = py[j] - cy, dz = pz[j] - cz;
            float d = dx * dx + dy * dy + dz * dz;
            dmin[j] = fminf(dmin[j], d);
            if (dmin[j] > bv) { bv = dmin[j]; bi = base + j; }
        }
        // wave argmax (prefer smaller index on ties)
#pragma unroll
        for (int off = 16; off >= 1; off >>= 1) {
            float ov = __shfl_xor(bv, off, 32);
            int   oi = __shfl_xor(bi, off, 32);
            if (ov > bv || (ov == bv && oi < bi)) { bv = ov; bi = oi; }
        }
        if (lane == 0) { rv[wid] = bv; ri[wid] = bi; }
        __syncthreads();
        if (wid == 0) {
            float v = rv[lane]; int i2 = ri[lane];
#pragma unroll
            for (int off = 16; off >= 1; off >>= 1) {
                float ov = __shfl_xor(v, off, 32);
                int   oi = __shfl_xor(i2, off, 32);
                if (ov > v || (ov == v && oi < i2)) { v = ov; i2 = oi; }
            }
            if (lane == 0) far_sh = i2;
        }
        __syncthreads();
        far = far_sh;
    }
}

// ---------------------------------------------------------------- ball query
// One wave per query point; stream-compact first NSAMPLE in-radius indices.
__global__ __launch_bounds__(256) void ballq_kernel(const float* __restrict__ xyz,
                                                    const float* __restrict__ newxyz,
                                                    int* __restrict__ gi) {
    const int q = blockIdx.x * 8 + (threadIdx.x >> 5);
    const int lane = threadIdx.x & 31;
    const int b = q / NPOINT, s = q % NPOINT;
    const float qx = newxyz[(b * 3 + 0) * NPOINT + s];
    const float qy = newxyz[(b * 3 + 1) * NPOINT + s];
    const float qz = newxyz[(b * 3 + 2) * NPOINT + s];
    const float r2 = RADIUS * RADIUS;
    int* out = gi + (size_t)(b * NPOINT + s) * NSAMPLE;

    int cnt = 0, firstIdx = -1;
    for (int base = 0; base < NQ; base += 32) {
        int i = base + lane;
        float dx = xyz[(b * 3 + 0) * NQ + i] - qx;
        float dy = xyz[(b * 3 + 1) * NQ + i] - qy;
        float dz = xyz[(b * 3 + 2) * NQ + i] - qz;
        bool in = (dx * dx + dy * dy + dz * dz) <= r2;
        unsigned mask = (unsigned)__ballot(in);
        if (firstIdx < 0 && mask) firstIdx = base + __ffs(mask) - 1;
        int before = __popc(mask & ((1u << lane) - 1u));
        if (in && (cnt + before) < NSAMPLE) out[cnt + before] = i;
        cnt += __popc(mask);
        if (cnt >= NSAMPLE) break;
    }
    if (cnt < NSAMPLE) {
        int pad = (firstIdx < 0) ? 0 : firstIdx;
        if (lane >= cnt && lane < NSAMPLE) out[lane] = pad;
    }
}

// ---------------------------------------------------------------- grouped MLP
// One block (512 thr = 16 waves) per (s, b). Tile: M=128 (8 mt) x N=32 (2 nt),
// K=68 (padded). STATS pass accumulates GN0 sums; APPLY pass does GN0+silu+max.
template <bool APPLY>
__global__ __launch_bounds__(512) void mlp_kernel(
    const float* __restrict__ xyz, const float* __restrict__ points,
    const float* __restrict__ newxyz, const int* __restrict__ gi,
    const float* __restrict__ w_mlp, const float* __restrict__ b_mlp,
    const float* __restrict__ g0, const float* __restrict__ be0,
    float* __restrict__ statsAcc, const float* __restrict__ statsFin,
    float* __restrict__ x_out) {

    __shared__ float Wlds[OUT_DIM * KPAD];   // 128 x 68
    __shared__ float Flds[NSAMPLE * KPAD];   // 32 x 68 (k-major)
    __shared__ float maxbuf[2 * OUT_DIM];

    const int s = blockIdx.x, b = blockIdx.y;
    const int tid = threadIdx.x;

    for (int e = tid; e < OUT_DIM * KPAD; e += 512) {
        int o = e / KPAD, c = e % KPAD;
        Wlds[e] = (c < IN_DIM + 3) ? w_mlp[o * (IN_DIM + 3) + c] : 0.0f;
    }
    const int* gis = gi + (size_t)(b * NPOINT + s) * NSAMPLE;
    for (int e = tid; e < NSAMPLE * KPAD; e += 512) {
        int k = e / KPAD, c = e % KPAD;
        int p = gis[k];
        float v;
        if (c < 3)
            v = xyz[(b * 3 + c) * NQ + p] - newxyz[(b * 3 + c) * NPOINT + s];
        else if (c < IN_DIM + 3)
            v = points[(b * IN_DIM + (c - 3)) * NQ + p];
        else
            v = 0.0f;
        Flds[e] = v;
    }
    __syncthreads();

    const int wave = tid >> 5, lane = tid & 31;
    const int mt = wave & 7, nt = wave >> 3;
    const int lm = lane & 15, kh = lane >> 4;

    AccU acc{};
    const float* Arow = &Wlds[(mt * 16 + lm) * KPAD];
    const float* Brow = &Flds[(nt * 16 + lm) * KPAD];
#pragma unroll
    for (int kk = 0; kk < KPAD; kk += 4)
        acc.v = wmma_step(Arow, Brow, kk, kh, acc.v);

    if (!APPLY) {
        float s1 = 0.0f, s2 = 0.0f;
#pragma unroll
        for (int v = 0; v < 8; ++v) {
            int ch = mt * 16 + v + 8 * kh;
            float val = acc.f[v] + b_mlp[ch];
            s1 += val; s2 += val * val;
        }
#pragma unroll
        for (int off = 16; off >= 1; off >>= 1) {
            s1 += __shfl_xor(s1, off, 32);
            s2 += __shfl_xor(s2, off, 32);
        }
        if (lane == 0) {
            atomicAdd(&statsAcc[((0 * BQ + b) * GROUPS + mt) * 2 + 0], s1);
            atomicAdd(&statsAcc[((0 * BQ + b) * GROUPS + mt) * 2 + 1], s2);
        }
    } else {
        const float mu = statsFin[((0 * BQ + b) * GROUPS + mt) * 2 + 0];
        const float is = statsFin[((0 * BQ + b) * GROUPS + mt) * 2 + 1];
        float rm[8];
#pragma unroll
        for (int v = 0; v < 8; ++v) {
            int ch = mt * 16 + v + 8 * kh;
            float val = acc.f[v] + b_mlp[ch];
            val = (val - mu) * is * g0[ch] + be0[ch];
            rm[v] = silu(val);
        }
#pragma unroll
        for (int off = 1; off <= 8; off <<= 1)
#pragma unroll
            for (int v = 0; v < 8; ++v)
                rm[v] = fmaxf(rm[v], __shfl_xor(rm[v], off, 32));
        if (lm == 0) {
#pragma unroll
            for (int v = 0; v < 8; ++v)
                maxbuf[nt * OUT_DIM + mt * 16 + v + 8 * kh] = rm[v];
        }
        __syncthreads();
        if (tid < OUT_DIM) {
            float m = fmaxf(maxbuf[tid], maxbuf[OUT_DIM + tid]);
            x_out[((size_t)b * OUT_DIM + tid) * NPOINT + s] = m;
        }
    }
}

// ---------------------------------------------------------------- h1 GEMM
// h1 = w1(512x128) @ x(128x2048) + b1; accumulate GN1 stats.
__global__ __launch_bounds__(512) void h1_kernel(const float* __restrict__ x,
                                                 const float* __restrict__ w1,
                                                 const float* __restrict__ b1,
                                                 float* __restrict__ h1,
                                                 float* __restrict__ statsAcc) {
    __shared__ float Alds[128 * 34];
    __shared__ float Blds[32 * 34];

    const int nblk = blockIdx.x * 32;
    const int mblk = blockIdx.y * 128;
    const int b = blockIdx.z;
    const int tid = threadIdx.x;
    const int wave = tid >> 5, lane = tid & 31;
    const int mt = wave & 7, nt = wave >> 3;
    const int lm = lane & 15, kh = lane >> 4;

    AccU acc{};
    for (int kc = 0; kc < 4; ++kc) {
        int k0 = kc * 32;
        __syncthreads();
        for (int e = tid; e < 128 * 32; e += 512) {
            int r = e >> 5, c = e & 31;
            Alds[r * 34 + c] = w1[(mblk + r) * OUT_DIM + k0 + c];
        }
        for (int e = tid; e < 32 * 32; e += 512) {
            int n = e & 31, c = e >> 5;
            Blds[n * 34 + c] =
                x[((size_t)b * OUT_DIM + k0 + c) * NPOINT + nblk + n];
        }
        __syncthreads();
        const float* Arow = &Alds[(mt * 16 + lm) * 34];
        const float* Brow = &Blds[(nt * 16 + lm) * 34];
#pragma unroll
        for (int kk = 0; kk < 32; kk += 4)
            acc.v = wmma_step(Arow, Brow, kk, kh, acc.v);
    }

    float s1 = 0.0f, s2 = 0.0f;
    const int n = nblk + nt * 16 + lm;
#pragma unroll
    for (int v = 0; v < 8; ++v) {
        int ch = mblk + mt * 16 + v + 8 * kh;
        float val = acc.f[v] + b1[ch];
        h1[((size_t)b * ED + ch) * NPOINT + n] = val;
        s1 += val; s2 += val * val;
    }
#pragma unroll
    for (int off = 16; off >= 1; off >>= 1) {
        s1 += __shfl_xor(s1, off, 32);
        s2 += __shfl_xor(s2, off, 32);
    }
    if (lane == 0) {
        int g = (mblk + mt * 16) / (ED / GROUPS);
        atomicAdd(&statsAcc[((1 * BQ + b) * GROUPS + g) * 2 + 0], s1);
        atomicAdd(&statsAcc[((1 * BQ + b) * GROUPS + g) * 2 + 1], s2);
    }
}

// ---------------------------------------------------------------- h2 GEMM
// h2 = w2(128x512) @ silu(gn1(h1)) + b2; accumulate GN2 stats.
__global__ __launch_bounds__(512) void h2_kernel(const float* __restrict__ h1,
                                                 const float* __restrict__ w2,
                                                 const float* __restrict__ b2,
                                                 const float* __restrict__ g1,
                                                 const float* __restrict__ be1,
                                                 const float* __restrict__ statsFin,
                                                 float* __restrict__ h2,
                                                 float* __restrict__ statsAcc) {
    __shared__ float Alds[128 * 34];
    __shared__ float Blds[32 * 34];

    const int nblk = blockIdx.x * 32;
    const int b = blockIdx.z;
    const int tid = threadIdx.x;
    const int wave = tid >> 5, lane = tid & 31;
    const int mt = wave & 7, nt = wave >> 3;
    const int lm = lane & 15, kh = lane >> 4;

    AccU acc{};
    for (int kc = 0; kc < 16; ++kc) {
        int k0 = kc * 32;
        __syncthreads();
        for (int e = tid; e < 128 * 32; e += 512) {
            int r = e >> 5, c = e & 31;
            Alds[r * 34 + c] = w2[r * ED + k0 + c];
        }
        for (int e = tid; e < 32 * 32; e += 512) {
            int n = e & 31, c = e >> 5;
            int ch1 = k0 + c;
            float raw = h1[((size_t)b * ED + ch1) * NPOINT + nblk + n];
            int gq = ch1 / (ED / GROUPS);
            float mu = statsFin[((1 * BQ + b) * GROUPS + gq) * 2 + 0];
            float is = statsFin[((1 * BQ + b) * GROUPS + gq) * 2 + 1];
            float val = (raw - mu) * is * g1[ch1] + be1[ch1];
            Blds[n * 34 + c] = silu(val);
        }
        __syncthreads();
        const float* Arow = &Alds[(mt * 16 + lm) * 34];
        const float* Brow = &Blds[(nt * 16 + lm) * 34];
#pragma unroll
        for (int kk = 0; kk < 32; kk += 4)
            acc.v = wmma_step(Arow, Brow, kk, kh, acc.v);
    }

    float s1 = 0.0f, s2 = 0.0f;
    const int n = nblk + nt * 16 + lm;
#pragma unroll
    for (int v = 0; v < 8; ++v) {
        int ch = mt * 16 + v + 8 * kh;
        float val = acc.f[v] + b2[ch];
        h2[((size_t)b * OUT_DIM + ch) * NPOINT + n] = val;
        s1 += val; s2 += val * val;
    }
#pragma unroll
    for (int off = 16; off >= 1; off >>= 1) {
        s1 += __shfl_xor(s1, off, 32);
        s2 += __shfl_xor(s2, off, 32);
    }
    if (lane == 0) {
        atomicAdd(&statsAcc[((2 * BQ + b) * GROUPS + mt) * 2 + 0], s1);
        atomicAdd(&statsAcc[((2 * BQ + b) * GROUPS + mt) * 2 + 1], s2);
    }
}

// ---------------------------------------------------------------- finalize GN
__global__ void finalize_kernel(const float* __restrict__ statsAcc,
                                float* __restrict__ statsFin,
                                int layer, float count) {
    int i = threadIdx.x;
    if (i < BQ * GROUPS) {
        float s1 = statsAcc[(layer * BQ * GROUPS + i) * 2 + 0];
        float s2 = statsAcc[(layer * BQ * GROUPS + i) * 2 + 1];
        float mu = s1 / count;
        float var = s2 / count - mu * mu;
        statsFin[(layer * BQ * GROUPS + i) * 2 + 0] = mu;
        statsFin[(layer * BQ * GROUPS + i) * 2 + 1] = rsqrtf(var + EPS);
    }
}

// ---------------------------------------------------------------- FiLM
__global__ void film_kernel(const float* __restrict__ t_emb,
                            const float* __restrict__ c_emb,
                            const float* __restrict__ wt, const float* __restrict__ bt,
                            const float* __restrict__ wc, const float* __restrict__ bc,
                            float* __restrict__ film) {
    const int b = blockIdx.x, j = threadIdx.x;  // j in [0,256)
    float at = bt[j], ac = bc[j];
    for (int c = 0; c < OUT_DIM; ++c) {
        at += t_emb[b * OUT_DIM + c] * wt[j * OUT_DIM + c];
        ac += c_emb[b * OUT_DIM + c] * wc[j * OUT_DIM + c];
    }
    float val = at + ac;
    if (j < OUT_DIM)
        film[(b * 2 + 0) * OUT_DIM + j] = val;           // scale
    else
        film[(b * 2 + 1) * OUT_DIM + (j - OUT_DIM)] = val;  // shift
}

// ---------------------------------------------------------------- final
__global__ void final_kernel(const float* __restrict__ h2,
                             const float* __restrict__ identity,
                             const float* __restrict__ statsFin,
                             const float* __restrict__ g2, const float* __restrict__ be2,
                             const float* __restrict__ film,
                             float* __restrict__ out_points) {
    size_t i = (size_t)blockIdx.x * blockDim.x + threadIdx.x;
    if (i >= (size_t)BQ * OUT_DIM * NPOINT) return;
    int b = (int)(i / (OUT_DIM * NPOINT));
    int ch = (int)((i / NPOINT) % OUT_DIM);
    int g = ch >> 4;
    float mu = statsFin[((2 * BQ + b) * GROUPS + g) * 2 + 0];
    float is = statsFin[((2 * BQ + b) * GROUPS + g) * 2 + 1];
    float h = (h2[i] - mu) * is * g2[ch] + be2[ch];
    float scale = film[(b * 2 + 0) * OUT_DIM + ch];
    float shift = film[(b * 2 + 1) * OUT_DIM + ch];
    out_points[i] = h * (1.0f + scale) + shift + identity[i];
}

// ================================================================ launcher
extern "C" void kernel_launch(void* const* d_in, const int* in_sizes, int n_in,
                              void* d_out, int out_size, void* d_ws, size_t ws_size,
                              hipStream_t stream) {
    const float* xyz    = (const float*)d_in[0];
    const float* points = (const float*)d_in[1];
    const float* t_emb  = (const float*)d_in[2];
    const float* c_emb  = (const float*)d_in[3];
    const float* w_mlp  = (const float*)d_in[4];
    const float* b_mlp  = (const float*)d_in[5];
    const float* g0     = (const float*)d_in[6];
    const float* be0    = (const float*)d_in[7];
    const float* w1     = (const float*)d_in[8];
    const float* b1     = (const float*)d_in[9];
    const float* g1     = (const float*)d_in[10];
    const float* be1    = (const float*)d_in[11];
    const float* w2     = (const float*)d_in[12];
    const float* b2     = (const float*)d_in[13];
    const float* g2     = (const float*)d_in[14];
    const float* be2    = (const float*)d_in[15];
    const float* wt     = (const float*)d_in[16];
    const float* bt     = (const float*)d_in[17];
    const float* wc     = (const float*)d_in[18];
    const float* bc     = (const float*)d_in[19];

    float* out_newxyz  = (float*)d_out;                               // B*3*NPOINT
    float* out_points  = (float*)d_out + (size_t)BQ * 3 * NPOINT;     // B*128*NPOINT

    // workspace carve-up (256B aligned)
    char* ws = (char*)d_ws;
    size_t off = 0;
    auto carve = [&](size_t bytes) {
        void* p = ws + off;
        off = (off + bytes + 255) & ~(size_t)255;
        return p;
    };
    int*   gi       = (int*)  carve((size_t)BQ * NPOINT * NSAMPLE * 4);
    float* xbuf     = (float*)carve((size_t)BQ * OUT_DIM * NPOINT * 4);
    float* h1buf    = (float*)carve((size_t)BQ * ED * NPOINT * 4);
    float* h2buf    = (float*)carve((size_t)BQ * OUT_DIM * NPOINT * 4);
    float* statsAcc = (float*)carve(3 * BQ * GROUPS * 2 * 4);
    float* statsFin = (float*)carve(3 * BQ * GROUPS * 2 * 4);
    float* film     = (float*)carve((size_t)BQ * 2 * OUT_DIM * 4);

    init_stats_kernel<<<1, 256, 0, stream>>>(statsAcc);

    fps_kernel<<<BQ, 1024, 0, stream>>>(xyz, out_newxyz);

    ballq_kernel<<<(BQ * NPOINT) / 8, 256, 0, stream>>>(xyz, out_newxyz, gi);

    dim3 mlpGrid(NPOINT, BQ);
    mlp_kernel<false><<<mlpGrid, 512, 0, stream>>>(
        xyz, points, out_newxyz, gi, w_mlp, b_mlp, g0, be0,
        statsAcc, statsFin, xbuf);
    finalize_kernel<<<1, 64, 0, stream>>>(statsAcc, statsFin, 0,
                                          (float)(16 * NPOINT * NSAMPLE));
    mlp_kernel<true><<<mlpGrid, 512, 0, stream>>>(
        xyz, points, out_newxyz, gi, w_mlp, b_mlp, g0, be0,
        statsAcc, statsFin, xbuf);

    h1_kernel<<<dim3(NPOINT / 32, ED / 128, BQ), 512, 0, stream>>>(
        xbuf, w1, b1, h1buf, statsAcc);
    finalize_kernel<<<1, 64, 0, stream>>>(statsAcc, statsFin, 1,
                                          (float)((ED / GROUPS) * NPOINT));

    h2_kernel<<<dim3(NPOINT / 32, 1, BQ), 512, 0, stream>>>(
        h1buf, w2, b2, g1, be1, statsFin, h2buf, statsAcc);
    finalize_kernel<<<1, 64, 0, stream>>>(statsAcc, statsFin, 2,
                                          (float)((OUT_DIM / GROUPS) * NPOINT));

    film_kernel<<<BQ, 256, 0, stream>>>(t_emb, c_emb, wt, bt, wc, bc, film);

    size_t total = (size_t)BQ * OUT_DIM * NPOINT;
    final_kernel<<<(unsigned)((total + 255) / 256), 256, 0, stream>>>(
        h2buf, xbuf, statsFin, g2, be2, film, out_points);
}